// AlignmentModel_86053964742721
// MI455X (gfx1250) — compile-verified
//
#include <hip/hip_runtime.h>

// -----------------------------------------------------------------------------
// Soft Needleman-Wunsch on MI455X (gfx1250, wave32).
//   Phase 1: theta = zx @ zy^T via V_WMMA_F32_16X16X4_F32 with 32x32 wave
//            tiles (2x2 register blocking -> 4 WMMAs per 4 fragment loads),
//            stored skewed: T[b][(i+j)&1023][i] so DP diagonals are contiguous.
//   Phase 2: per-batch gap scalar A[b] (tiny reduction).
//   Phase 3: wavefront DP, one 1024-thread workgroup per batch, 3 rotating
//            diagonal buffers in LDS, one barrier per anti-diagonal, with
//            global_prefetch of the next diagonal's theta row.
// -----------------------------------------------------------------------------

typedef __attribute__((ext_vector_type(2))) float v2f;
typedef __attribute__((ext_vector_type(8))) float v8f;

#define NEGV (-1000000000.0f)

static __device__ __forceinline__ v8f wmma_f32(v2f a, v2f b, v8f c) {
  // 8 args: (neg_a, A, neg_b, B, c_mod, C, reuse_a, reuse_b)
  return __builtin_amdgcn_wmma_f32_16x16x4_f32(false, a, false, b,
                                               (short)0, c, false, false);
}

// Store one 16x16 f32 C/D tile into the skewed theta layout.
// C/D layout: VGPR v, lanes 0-15 -> M = v, N = lane; lanes 16-31 -> M = v+8.
static __device__ __forceinline__ void store_tile(float* __restrict__ Tb,
                                                  v8f c, int mi, int nj,
                                                  int r, int half) {
  const int n = nj + r;
#pragma unroll
  for (int v = 0; v < 8; ++v) {
    const int m = mi + v + 8 * half;
    const int cidx = (m + n) & 1023;               // skewed column
    Tb[((size_t)cidx << 10) + m] = c[v];
  }
}

// ---------------------------------------------------------------- Phase 1 ----
// One wave per 32x32 output tile (4 accumulators). K looped in steps of 4.
// A-frag layout (16x4 f32): lanes 0-15 hold M=lane, VGPR{0,1}=K{0,1};
//                           lanes 16-31 hold M=lane-16, VGPR{0,1}=K{2,3}.
// B-frag (4x16, B = zy^T) is symmetric in N -> identical row-major loads.
__global__ __launch_bounds__(256)
void theta_gemm_kernel(const float* __restrict__ zx,
                       const float* __restrict__ zy,
                       float* __restrict__ thetaS) {
  const int wave = threadIdx.x >> 5;
  const int lane = threadIdx.x & 31;
  const int r    = lane & 15;
  const int half = lane >> 4;

  const int id  = blockIdx.x * 8 + wave;    // 0 .. 16383 supertiles
  const int b   = id >> 10;                 // 16 batches
  const int rem = id & 1023;
  const int ti  = (rem >> 5) * 32;          // 32 row supertiles
  const int tj  = (rem & 31) * 32;          // 32 col supertiles

  const float* __restrict__ A0 = zx + (((size_t)b * 1024 + ti + r) * 512);
  const float* __restrict__ A1 = A0 + (size_t)16 * 512;
  const float* __restrict__ B0 = zy + (((size_t)b * 1024 + tj + r) * 512);
  const float* __restrict__ B1 = B0 + (size_t)16 * 512;
  const int ko = 2 * half;

  v8f c00 = {}, c01 = {}, c10 = {}, c11 = {};
#pragma unroll 2
  for (int k = 0; k < 512; k += 4) {
    v2f a0 = *(const v2f*)(A0 + k + ko);
    v2f a1 = *(const v2f*)(A1 + k + ko);
    v2f b0 = *(const v2f*)(B0 + k + ko);
    v2f b1 = *(const v2f*)(B1 + k + ko);
    c00 = wmma_f32(a0, b0, c00);
    c01 = wmma_f32(a0, b1, c01);
    c10 = wmma_f32(a1, b0, c10);
    c11 = wmma_f32(a1, b1, c11);
  }

  float* __restrict__ Tb = thetaS + ((size_t)b << 20);
  store_tile(Tb, c00, ti,      tj,      r, half);
  store_tile(Tb, c01, ti,      tj + 16, r, half);
  store_tile(Tb, c10, ti + 16, tj,      r, half);
  store_tile(Tb, c11, ti + 16, tj + 16, r, half);
}

// ---------------------------------------------------------------- Phase 2 ----
// A[b] = (sum_{i,d} zx[b,i,d]*w[d] + sum_{j,d} zy[b,j,d]*w[512+d]) / 1024 + gb
__global__ __launch_bounds__(256)
void gap_kernel(const float* __restrict__ zx,
                const float* __restrict__ zy,
                const float* __restrict__ gw,
                const float* __restrict__ gb,
                float* __restrict__ Agap) {
  const int b = blockIdx.x;
  const float* __restrict__ X = zx + (size_t)b * 1024 * 512;
  const float* __restrict__ Y = zy + (size_t)b * 1024 * 512;

  float acc = 0.0f;
  for (int idx = threadIdx.x; idx < 1024 * 512; idx += 256) {
    const int dd = idx & 511;
    acc += X[idx] * gw[dd] + Y[idx] * gw[512 + dd];
  }

  __shared__ float red[256];
  red[threadIdx.x] = acc;
  __syncthreads();
#pragma unroll
  for (int s = 128; s > 0; s >>= 1) {
    if ((int)threadIdx.x < s) red[threadIdx.x] += red[threadIdx.x + s];
    __syncthreads();
  }
  if (threadIdx.x == 0) Agap[b] = red[0] * (1.0f / 1024.0f) + gb[0];
}

// ---------------------------------------------------------------- Phase 3 ----
// One block per batch. Diagonal vectors (length 1025, indexed by row i) kept
// in 3 rotating LDS buffers. newvals[i] (stored at slot i+1) depends on
// p1[i] (up), p2[i] (diag), p1[i+1] (left); theta row for diagonal d is the
// contiguous skewed row T[b][(d-2)&1023][*]. The next diagonal's theta row is
// prefetched (global_prefetch_b8) so its L2 latency hides behind the barrier.
__global__ __launch_bounds__(1024)
void nw_dp_kernel(const float* __restrict__ thetaS,
                  const float* __restrict__ Agap,
                  float* __restrict__ out) {
  __shared__ float buf[3][1025];
  const int b = blockIdx.x;
  const int tid = threadIdx.x;          // row index ii = tid, 0..1023
  const float A = Agap[b];
  const float* __restrict__ Tb = thetaS + ((size_t)b << 20);

  // prev2[0]=0 else NEG; prev all NEG
  buf[0][tid] = (tid == 0) ? 0.0f : NEGV;
  buf[1][tid] = NEGV;
  if (tid == 0) { buf[0][1024] = NEGV; buf[1][1024] = NEGV; }
  // warm L2 with the first diagonal's theta row
  __builtin_prefetch(Tb + tid, 0, 1);
  __syncthreads();

  int i2 = 0, i1 = 1, i0 = 2;
  for (int d = 2; d <= 2048; ++d) {
    const int jj = d - tid - 2;
    const float up = buf[i1][tid];
    const float dg = buf[i2][tid];
    const float lf = buf[i1][tid + 1];

    // prefetch theta row for diagonal d+1 (harmless overshoot at d==2048)
    __builtin_prefetch(Tb + (((size_t)((d - 1) & 1023)) << 10) + tid, 0, 1);

    float nv = NEGV;
    if (jj >= 0 && jj < 1024) {
      const float th = Tb[(((size_t)(d - 2) & 1023) << 10) + tid];
      const float ua = up + A;
      const float la = lf + A;
      const float m  = fmaxf(ua, fmaxf(dg, la));
      const float s  = __expf(ua - m) + __expf(dg - m) + __expf(la - m);
      nv = th + m + __logf(s);
    }
    buf[i0][tid + 1] = nv;
    if (tid == 0) buf[i0][0] = NEGV;
    __syncthreads();
    const int t = i2; i2 = i1; i1 = i0; i0 = t;   // rotate (uniform)
  }

  if (tid == 0) out[b] = buf[i1][1024];           // V[N, M]
}

// ---------------------------------------------------------------- launch ----
extern "C" void kernel_launch(void* const* d_in, const int* in_sizes, int n_in,
                              void* d_out, int out_size, void* d_ws, size_t ws_size,
                              hipStream_t stream) {
  const float* zx = (const float*)d_in[0];   // [16,1024,512]
  const float* zy = (const float*)d_in[1];   // [16,1024,512]
  const float* gw = (const float*)d_in[2];   // [1,1024]
  const float* gb = (const float*)d_in[3];   // [1]

  float* thetaS = (float*)d_ws;                    // 16*1024*1024 f32 (64 MB)
  float* Agap   = thetaS + ((size_t)16 << 20);     // 16 f32

  // 16384 wave-tiles (32x32 each) / 8 waves per 256-thread block
  theta_gemm_kernel<<<2048, 256, 0, stream>>>(zx, zy, thetaS);
  gap_kernel<<<16, 256, 0, stream>>>(zx, zy, gw, gb, Agap);
  nw_dp_kernel<<<16, 1024, 0, stream>>>(thetaS, Agap, (float*)d_out);
}